// YoloLoss_82274393523046
// MI455X (gfx1250) — compile-verified
//
#include <hip/hip_runtime.h>
#include <hip/hip_bf16.h>

// ---------------- problem constants ----------------
#define BB 8
#define AA 3
#define CC 80
#define TT 16
#define MAPW 128
#define MAPH 128
#define ATTRS (5 + CC)          // 85
#define PLANE (MAPW * MAPH)     // 16384 floats per (b, channel) plane
#define NCELLBLK (BB * AA * MAPW)  // 3072 blocks for the dense pass

typedef __attribute__((ext_vector_type(2))) float v2f;
typedef __attribute__((ext_vector_type(8))) float v8f;

// jnp.nan_to_num defaults: nan->0, +inf->FLT_MAX, -inf->-FLT_MAX (branchless)
__device__ __forceinline__ float fixv(float v) {
    v = (v != v) ? 0.0f : v;
    v = fminf(v, 3.4028234663852886e38f);
    v = fmaxf(v, -3.4028234663852886e38f);
    return v;
}

// Exact f32 sum of all 32 lanes using V_WMMA_F32_16X16X4_F32.
// A (16x4): lane L holds A[L%16][L<16?0:2] in vgpr0 (our value), vgpr1 = 0.
// B (4x16): all ones (layout independent). D[m][n] = v[m] + v[m+16].
// Per lane, sum of the 8 D vgprs = half-wave sum; xor-16 shuffle completes it.
__device__ __forceinline__ float waveReduceWmma(float v) {
    v2f a; a[0] = v;    a[1] = 0.0f;
    v2f b; b[0] = 1.0f; b[1] = 1.0f;
    v8f c = {};
    c = __builtin_amdgcn_wmma_f32_16x16x4_f32(
            /*neg_a=*/false, a, /*neg_b=*/false, b,
            /*c_mod=*/(short)0, c, /*reuse_a=*/false, /*reuse_b=*/false);
    float s = ((c[0] + c[1]) + (c[2] + c[3])) + ((c[4] + c[5]) + (c[6] + c[7]));
    s += __shfl_xor(s, 16, 32);
    return s;  // total in every lane
}

// ---------------- kernel 1: per-target prep (1 block, B*T=128 threads) -----
__global__ void prep_kernel(const float* __restrict__ boxes,
                            const float* __restrict__ areas,
                            const float* __restrict__ anchors,
                            float* __restrict__ tb,   // [B*T*5] cx,cy,w,h,areaT
                            int* __restrict__ win) {  // [B*T*4] winner,a,gx,gy
    const int i = threadIdx.x;     // 0..127
    const int b = i / TT, t = i % TT;
    (void)t;
    const float x1 = boxes[i * 4 + 0], y1 = boxes[i * 4 + 1];
    const float x2 = boxes[i * 4 + 2], y2 = boxes[i * 4 + 3];
    const float cx = (x1 + x2) * 0.5f, cy = (y1 + y2) * 0.5f;
    const float w = x2 - x1, h = y2 - y1;
    const float area = areas[i];

    const float aA0 = anchors[0] * anchors[1];
    const float aA1 = anchors[2] * anchors[3];
    const float aA2 = anchors[4] * anchors[5];
    int bestA = 0;
    float bd = fabsf(aA0 - area);
    float d1 = fabsf(aA1 - area); if (d1 < bd) { bd = d1; bestA = 1; }
    float d2 = fabsf(aA2 - area); if (d2 < bd) { bd = d2; bestA = 2; }

    // strideW = strideH = 1024/128 = 8 exactly
    int gx = (int)floorf((w - cx) * 0.125f); gx = min(max(gx, 0), MAPW - 1);
    int gy = (int)floorf((h - cy) * 0.125f); gy = min(max(gy, 0), MAPH - 1);

    tb[i * 5 + 0] = cx;
    tb[i * 5 + 1] = cy;
    tb[i * 5 + 2] = w;
    tb[i * 5 + 3] = h;
    tb[i * 5 + 4] = (w - cx) * (h - cy);   // reference's areaT quirk

    __shared__ int cell[BB * TT];
    cell[i] = (bestA << 14) | (gx << 7) | gy;
    __syncthreads();

    // sequential .at[].set(t): largest t touching a cell wins
    int winner = 1;
    const int myCell = cell[i];
    for (int tp = (i % TT) + 1; tp < TT; ++tp)
        if (cell[b * TT + tp] == myCell) { winner = 0; }

    win[i * 4 + 0] = winner;
    win[i * 4 + 1] = bestA;
    win[i * 4 + 2] = gx;
    win[i * 4 + 3] = gy;
}

// ---------------- kernel 2: dense IoU / objLoss partials -------------------
// grid = B*A*MAPW blocks, block = MAPH(=128) threads; one y-row per block.
__global__ void iou_kernel(const float* __restrict__ outp,
                           const float* __restrict__ tb,
                           float* __restrict__ blkObj,
                           float* __restrict__ blkLow) {
    const int blk = blockIdx.x;
    const int b = blk / (AA * MAPW);
    const int rem = blk % (AA * MAPW);
    const int a = rem / MAPW;
    const int x = rem % MAPW;
    const int y = threadIdx.x;

    __shared__ float s_tb[TT * 5];
    if (threadIdx.x < TT * 5) s_tb[threadIdx.x] = tb[b * (TT * 5) + threadIdx.x];
    __syncthreads();

    const int base = ((b * (AA * ATTRS) + a * ATTRS) * MAPW + x) * MAPH + y;
    const float pcx = fixv(outp[base + 0 * PLANE]);
    const float pcy = fixv(outp[base + 1 * PLANE]);
    const float pw  = fixv(outp[base + 2 * PLANE]);
    const float ph  = fixv(outp[base + 3 * PLANE]);

    const float p1x = pcx - pw * 0.5f, p1y = pcy - ph * 0.5f;
    const float p2x = pcx + pw * 0.5f, p2y = pcy + ph * 0.5f;
    const float areaP = (p2x - p1x) * (p2y - p1y);

    float best = -INFINITY;
#pragma unroll
    for (int t = 0; t < TT; ++t) {
        const float t1x = s_tb[t * 5 + 0], t1y = s_tb[t * 5 + 1];
        const float t2x = s_tb[t * 5 + 2], t2y = s_tb[t * 5 + 3];
        const float aT  = s_tb[t * 5 + 4];
        const float ix1 = fmaxf(p1x, t1x), iy1 = fmaxf(p1y, t1y);
        const float ix2 = fminf(p2x, t2x), iy2 = fminf(p2y, t2y);
        const float inter = fmaxf(ix2 - ix1, 0.0f) * fmaxf(iy2 - iy1, 0.0f);
        const float iou = inter / (areaP + aT - inter + 1e-16f);
        best = fmaxf(best, iou);
    }
    const float low = (best < 0.5f) ? 1.0f : 0.0f;
    const float obj = low * (1.0f - best);

    // deterministic block reduction: WMMA wave sums -> LDS -> thread 0
    const float so = waveReduceWmma(obj);
    const float sl = waveReduceWmma(low);
    __shared__ float sO[4], sL[4];
    const int wid = threadIdx.x >> 5, lane = threadIdx.x & 31;
    if (lane == 0) { sO[wid] = so; sL[wid] = sl; }
    __syncthreads();
    if (threadIdx.x == 0) {
        blkObj[blk] = (sO[0] + sO[1]) + (sO[2] + sO[3]);
        blkLow[blk] = (sL[0] + sL[1]) + (sL[2] + sL[3]);
    }
}

// ---------------- kernel 3: final reduction + sparse box/cls ---------------
__global__ void final_kernel(const float* __restrict__ outp,
                             const int* __restrict__ labels,
                             const float* __restrict__ tb,
                             const int* __restrict__ win,
                             const float* __restrict__ blkObj,
                             const float* __restrict__ blkLow,
                             float* __restrict__ out4) {
    const int tid = threadIdx.x;  // 256 threads

    // fixed-order strided accumulation of dense partials (deterministic)
    float myObj = 0.0f, myLow = 0.0f;
    for (int i = tid; i < NCELLBLK; i += 256) {
        myObj += blkObj[i];
        myLow += blkLow[i];
    }

    // sparse winner terms (tid < 128 maps to (b,t) like prep_kernel)
    float boxv = 0.0f, clsv = 0.0f;
    if (tid < BB * TT) {
        if (win[tid * 4 + 0]) {
            const int b = tid / TT;
            const int a = win[tid * 4 + 1];
            const int gx = win[tid * 4 + 2];
            const int gy = win[tid * 4 + 3];
            const int base = ((b * (AA * ATTRS) + a * ATTRS) * MAPW + gx) * MAPH + gy;
#pragma unroll
            for (int c = 0; c < 4; ++c) {
                const float d = fixv(outp[base + c * PLANE]) - tb[tid * 5 + c];
                boxv += d * d;
            }
            const int lab = labels[tid] - 1;
            for (int c = 0; c < CC; ++c) {
                const float l = fixv(outp[base + (5 + c) * PLANE]);
                // jax.nn.softplus(x) = max(x,0) + log1p(exp(-|x|))
                const float sp = fmaxf(l, 0.0f) + log1pf(expf(-fabsf(l)));
                clsv += sp - l * ((c == lab) ? 1.0f : 0.0f);
            }
        }
    }

    const float so = waveReduceWmma(myObj);
    const float sl = waveReduceWmma(myLow);
    const float sb = waveReduceWmma(boxv);
    const float sc = waveReduceWmma(clsv);

    __shared__ float red[4 * 8];
    const int wid = tid >> 5, lane = tid & 31;
    if (lane == 0) {
        red[wid] = so; red[8 + wid] = sl; red[16 + wid] = sb; red[24 + wid] = sc;
    }
    __syncthreads();
    if (tid == 0) {
        float O = 0.0f, L = 0.0f, Bx = 0.0f, Cs = 0.0f;
        for (int w = 0; w < 8; ++w) {
            O += red[w]; L += red[8 + w]; Bx += red[16 + w]; Cs += red[24 + w];
        }
        const float objLoss = 0.05f * O / L;
        const float clsLoss = 0.5f * Cs;
        out4[0] = Bx + objLoss + clsLoss;  // totalLoss
        out4[1] = Bx;                      // boxLoss
        out4[2] = objLoss;                 // objLoss
        out4[3] = clsLoss;                 // clsLoss
    }
}

extern "C" void kernel_launch(void* const* d_in, const int* in_sizes, int n_in,
                              void* d_out, int out_size, void* d_ws, size_t ws_size,
                              hipStream_t stream) {
    (void)in_sizes; (void)n_in; (void)out_size; (void)ws_size;
    const float* outp    = (const float*)d_in[0];
    const float* boxes   = (const float*)d_in[1];
    const int*   labels  = (const int*)d_in[2];
    const float* areas   = (const float*)d_in[3];
    const float* anchors = (const float*)d_in[4];

    // workspace layout (floats): [0,640) tb | [640,1152) win(int) |
    // [2048,5120) blkObj | [5120,8192) blkLow   -> 32 KB total
    float* ws_f  = (float*)d_ws;
    float* tb    = ws_f;
    int*   win   = (int*)(ws_f + 640);
    float* blkObj = ws_f + 2048;
    float* blkLow = ws_f + 2048 + NCELLBLK;

    prep_kernel<<<1, BB * TT, 0, stream>>>(boxes, areas, anchors, tb, win);
    iou_kernel<<<NCELLBLK, MAPH, 0, stream>>>(outp, tb, blkObj, blkLow);
    final_kernel<<<1, 256, 0, stream>>>(outp, labels, tb, win, blkObj, blkLow,
                                        (float*)d_out);
}